// EdgeConv_10934986735968
// MI455X (gfx1250) — compile-verified
//
#include <hip/hip_runtime.h>
#include <hip/hip_bf16.h>
#include <float.h>

#define B_    8
#define CIN   64
#define NPTS  4096
#define COUT  64
#define KNN   20
#define EPSV  1e-5f
#define NEGS  0.2f

typedef __attribute__((ext_vector_type(2))) float v2f;
typedef __attribute__((ext_vector_type(8))) float v8f;

// ---------------------------------------------------------------------------
// Kernel 1: per-point squared norms  norms[b*N+n] = sum_c x[b][c][n]^2
// ---------------------------------------------------------------------------
__global__ __launch_bounds__(256) void norms_kernel(const float* __restrict__ x,
                                                    float* __restrict__ norms) {
    int i = blockIdx.x * 256 + threadIdx.x;          // i in [0, B*N)
    int b = i >> 12;
    int n = i & (NPTS - 1);
    const float* xp = x + (size_t)b * CIN * NPTS + n;
    float s = 0.f;
#pragma unroll 8
    for (int c = 0; c < CIN; ++c) {
        float v = xp[(size_t)c * NPTS];
        s += v * v;
    }
    norms[i] = s;
}

// ---------------------------------------------------------------------------
// Kernel 2: kNN via WMMA Gram matrix.  One wave32 per 16-row tile.
// A/B fragment layout for V_WMMA_F32_16X16X4_F32 (32-bit, 16x4 / 4x16):
//   lanes 0-15  : M/N = lane,    K = 4*kk+0 (v0), 4*kk+1 (v1)
//   lanes 16-31 : M/N = lane-16, K = 4*kk+2 (v0), 4*kk+3 (v1)
// C/D layout: VGPR j -> row M = j + 8*(lane>=16), col N = lane&15
// ---------------------------------------------------------------------------
__global__ __launch_bounds__(32) void knn_kernel(const float* __restrict__ x,
                                                 const float* __restrict__ norms,
                                                 int* __restrict__ knn_idx) {
    __shared__ float tile[16][17];

    int blk  = blockIdx.x;            // B * (N/16)
    int b    = blk >> 8;
    int n0   = (blk & 255) * 16;
    int lane = threadIdx.x;
    int m    = lane & 15;
    int hi   = lane >> 4;

    const float* xb = x + (size_t)b * CIN * NPTS;
    const float* nb = norms + (size_t)b * NPTS;

    // Preload A fragments for this block's 16 rows (points n0..n0+15).
    v2f A[16];
#pragma unroll
    for (int kk = 0; kk < 16; ++kk) {
        int ch = kk * 4 + hi * 2;
        A[kk].x = xb[(size_t)ch * NPTS + n0 + m];
        A[kk].y = xb[(size_t)(ch + 1) * NPTS + n0 + m];
    }

    // Row norms for each accumulator register j.
    float rn[8];
#pragma unroll
    for (int j = 0; j < 8; ++j) rn[j] = nb[n0 + j + hi * 8];

    // Register-resident sorted (ascending) top-K list, kd[K-1] = current max.
    float kd[KNN];
    int   ki[KNN];
#pragma unroll
    for (int k = 0; k < KNN; ++k) { kd[k] = FLT_MAX; ki[k] = 0; }

    for (int ct = 0; ct < NPTS / 16; ++ct) {
        int m0 = ct * 16;
        v8f acc = {0.f, 0.f, 0.f, 0.f, 0.f, 0.f, 0.f, 0.f};
#pragma unroll
        for (int kk = 0; kk < 16; ++kk) {
            int ch = kk * 4 + hi * 2;
            v2f Bf;
            Bf.x = xb[(size_t)ch * NPTS + m0 + m];
            Bf.y = xb[(size_t)(ch + 1) * NPTS + m0 + m];
            acc = __builtin_amdgcn_wmma_f32_16x16x4_f32(
                false, A[kk], false, Bf, (short)0, acc, false, false);
        }
        float cn = nb[m0 + m];
#pragma unroll
        for (int j = 0; j < 8; ++j) {
            tile[j + hi * 8][m] = rn[j] + cn - 2.0f * acc[j];
        }
        __syncthreads();
        if (lane < 16) {
#pragma unroll 4
            for (int c = 0; c < 16; ++c) {
                float d = tile[lane][c];
                if (d < kd[KNN - 1]) {
                    kd[KNN - 1] = d;
                    ki[KNN - 1] = m0 + c;
#pragma unroll
                    for (int j = KNN - 1; j > 0; --j) {
                        if (kd[j] < kd[j - 1]) {
                            float td = kd[j]; kd[j] = kd[j - 1]; kd[j - 1] = td;
                            int   ti = ki[j]; ki[j] = ki[j - 1]; ki[j - 1] = ti;
                        }
                    }
                }
            }
        }
        __syncthreads();
    }

    if (lane < 16) {
        int* op = knn_idx + ((size_t)b * NPTS + n0 + lane) * KNN;
#pragma unroll
        for (int k = 0; k < KNN; ++k) op[k] = ki[k];
    }
}

// ---------------------------------------------------------------------------
// Kernel 3: edge-feature GEMM (WMMA) + BN + LeakyReLU + max over K.
// One block = 4 points -> 80 (point,k) rows = 5 M-tiles; 4 N-tiles of C_OUT.
// ---------------------------------------------------------------------------
__global__ __launch_bounds__(256) void edge_kernel(const float* __restrict__ x,
                                                   const float* __restrict__ W,
                                                   const float* __restrict__ gamma,
                                                   const float* __restrict__ beta,
                                                   const float* __restrict__ rmean,
                                                   const float* __restrict__ rvar,
                                                   const int* __restrict__ knn_idx,
                                                   float* __restrict__ out) {
    __shared__ float ef[80][129];   // edge features, padded vs 64 banks
    __shared__ float hl[80][66];    // post-BN activations
    __shared__ int   nbr[80];

    int tid = threadIdx.x;
    int blk = blockIdx.x;           // B * (N/4)
    int b   = blk >> 10;
    int n0  = (blk & 1023) * 4;

    const float* xb = x + (size_t)b * CIN * NPTS;

    if (tid < 80) {
        int p = tid / KNN;
        int k = tid - p * KNN;
        nbr[tid] = knn_idx[((size_t)b * NPTS + n0 + p) * KNN + k];
    }
    __syncthreads();

    // Build 80 x 128 edge matrix: [central(64) | neighbor-central(64)]
    for (int e = tid; e < 80 * 128; e += 256) {
        int r = e >> 7;
        int c = e & 127;
        int p = r / KNN;
        int n = n0 + p;
        float v;
        if (c < CIN) {
            v = xb[(size_t)c * NPTS + n];
        } else {
            int cc = c - CIN;
            v = xb[(size_t)cc * NPTS + nbr[r]] - xb[(size_t)cc * NPTS + n];
        }
        ef[r][c] = v;
    }
    __syncthreads();

    int w    = tid >> 5;
    int lane = tid & 31;
    int m    = lane & 15;
    int hi   = lane >> 4;

    for (int u = w; u < 20; u += 8) {        // 5 M-tiles x 4 N-tiles
        int mt = u >> 2;
        int nt = u & 3;
        int r  = mt * 16 + m;                // (point,k) row, 0..79
        int o  = nt * 16 + m;                // output channel
        const float* wrow = W + (size_t)o * 128 + hi * 2;

        v8f acc = {0.f, 0.f, 0.f, 0.f, 0.f, 0.f, 0.f, 0.f};
#pragma unroll 4
        for (int kk = 0; kk < 32; ++kk) {
            int ch = kk * 4 + hi * 2;
            v2f a;
            a.x = ef[r][ch];
            a.y = ef[r][ch + 1];
            v2f bf;
            bf.x = wrow[kk * 4];
            bf.y = wrow[kk * 4 + 1];
            acc = __builtin_amdgcn_wmma_f32_16x16x4_f32(
                false, a, false, bf, (short)0, acc, false, false);
        }

        float sc = rsqrtf(rvar[o] + EPSV) * gamma[o];
        float sh = beta[o] - rmean[o] * sc;
#pragma unroll
        for (int j = 0; j < 8; ++j) {
            float h = acc[j] * sc + sh;
            h = (h >= 0.f) ? h : NEGS * h;
            hl[mt * 16 + hi * 8 + j][o] = h;
        }
    }
    __syncthreads();

    // max over K for 4 points x 64 channels = 256 threads
    {
        int p = tid >> 6;
        int c = tid & 63;
        float mx = -FLT_MAX;
#pragma unroll
        for (int k = 0; k < KNN; ++k) mx = fmaxf(mx, hl[p * KNN + k][c]);
        out[((size_t)b * COUT + c) * NPTS + n0 + p] = mx;
    }
}

// ---------------------------------------------------------------------------
extern "C" void kernel_launch(void* const* d_in, const int* in_sizes, int n_in,
                              void* d_out, int out_size, void* d_ws, size_t ws_size,
                              hipStream_t stream) {
    (void)in_sizes; (void)n_in; (void)out_size; (void)ws_size;

    const float* x     = (const float*)d_in[0];
    const float* W     = (const float*)d_in[1];
    const float* gamma = (const float*)d_in[2];
    const float* beta  = (const float*)d_in[3];
    const float* rmean = (const float*)d_in[4];
    const float* rvar  = (const float*)d_in[5];
    float* out = (float*)d_out;

    float* norms   = (float*)d_ws;
    int*   knn_idx = (int*)((char*)d_ws + (size_t)B_ * NPTS * sizeof(float));

    norms_kernel<<<(B_ * NPTS) / 256, 256, 0, stream>>>(x, norms);
    knn_kernel<<<B_ * (NPTS / 16), 32, 0, stream>>>(x, norms, knn_idx);
    edge_kernel<<<B_ * (NPTS / 4), 256, 0, stream>>>(x, W, gamma, beta, rmean,
                                                     rvar, knn_idx, out);
}